// BlurTensor_37263136260832
// MI455X (gfx1250) — compile-verified
//
#include <hip/hip_runtime.h>
#include <hip/hip_bf16.h>

typedef __attribute__((ext_vector_type(2))) float v2f;
typedef __attribute__((ext_vector_type(8))) float v8f;

#define RADIUS   20
#define KLEN     41            // 2*RADIUS+1
#define KTOT     56            // KLEN + 15  (K window for a 16-wide output tile)
#define KSTEPS   14            // KTOT / 4
#define PADW     552           // 512 + 2*RADIUS
#define LDSP     554           // LDS row pitch (floats): even (b64 align), 554%64=42 -> conflict-free
#define WPAD_OFF 15
#define WPAD_LEN 72            // covers w indices [-15, 55] with zero padding
#define IMG      (512 * 512)
#define CSTR     IMG
#define NSTR     (3 * IMG)

__device__ __forceinline__ int reflect512(int j) {
    if (j < 0) j = -1 - j;
    else if (j >= 512) j = 1023 - j;
    return j;
}

// full symmetric-reflect fold for small axes (multiple wraps)
__device__ __forceinline__ int reflect_gen(int j, int n) {
    int p = j % (2 * n);
    if (p < 0) p += 2 * n;
    return (p < n) ? p : (2 * n - 1 - p);
}

// ---------------------------------------------------------------------------
// Heavy passes: 1-D Gaussian along W (ALONG_H=false) or H (ALONG_H=true)
// via V_WMMA_F32_16X16X4_F32. One workgroup = 16 rows(/cols) x full 512 axis.
// ---------------------------------------------------------------------------
template <bool ALONG_H>
__global__ __launch_bounds__(256) void blur512_wmma(const float* __restrict__ in,
                                                    float* __restrict__ out,
                                                    const int* __restrict__ k_size) {
    __shared__ float s_tile[16 * LDSP];   // 16 rows x (512 + halo) padded strip
    __shared__ float s_wpad[WPAD_LEN];    // zero-padded gaussian taps
    __shared__ float s_norm;

    const int tid = threadIdx.x;
    const float sigma = (float)k_size[0];

    // ---- build normalized gaussian, zero-padded for banded-matrix reads ----
    if (tid < WPAD_LEN) s_wpad[tid] = 0.0f;
    __syncthreads();
    if (tid < KLEN) {
        float t = (float)(tid - RADIUS) / sigma;
        s_wpad[WPAD_OFF + tid] = __expf(-0.5f * t * t);
    }
    __syncthreads();
    if (tid == 0) {
        float s = 0.0f;
        for (int k = 0; k < KLEN; ++k) s += s_wpad[WPAD_OFF + k];
        s_norm = 1.0f / s;
    }
    __syncthreads();
    if (tid < KLEN) s_wpad[WPAD_OFF + tid] *= s_norm;

    // ---- stage input strip (with symmetric halo) into LDS, coalesced ----
    if (!ALONG_H) {
        // 16 consecutive flat rows (n,c,h), conv along contiguous w
        const float* src = in + (size_t)blockIdx.x * (16 * 512);
        const int m = tid >> 4, j0 = tid & 15;
        for (int j = j0; j < PADW; j += 16)
            s_tile[m * LDSP + j] = src[m * 512 + reflect512(j - RADIUS)];
    } else {
        // 16 consecutive w columns of one (n,c) image, conv along h
        const float* src = in + (size_t)blockIdx.y * IMG + blockIdx.x * 16;
        const int mw = tid & 15;
        for (int j = tid >> 4; j < PADW; j += 16)
            s_tile[mw * LDSP + j] = src[(size_t)reflect512(j - RADIUS) * 512 + mw];
    }
    __syncthreads();

    // ---- per-wave 16x16 output tiles, D = data x band (or band x data) ----
    const int lane = tid & 31;
    const int wv   = tid >> 5;       // 8 waves -> 4 column tiles each
    const int idx  = lane & 15;      // M (or N) coordinate of this lane
    const int half = lane >> 4;      // K-half select per 16x4 / 4x16 layout
    const int kk0  = 2 * half;

    // banded gaussian fragments: band[k][idx] = w[k - idx]  (zero-padded)
    v2f wf[KSTEPS];
#pragma unroll
    for (int ks = 0; ks < KSTEPS; ++ks) {
        const int k = ks * 4 + kk0;
        wf[ks].x = s_wpad[WPAD_OFF + k - idx];
        wf[ks].y = s_wpad[WPAD_OFF + k + 1 - idx];
    }

    const float* arow = &s_tile[idx * LDSP];

    for (int t = 0; t < 4; ++t) {
        const int c0 = (wv * 4 + t) * 16;    // tile base along the conv axis
        v8f acc = {0.f, 0.f, 0.f, 0.f, 0.f, 0.f, 0.f, 0.f};
#pragma unroll
        for (int ks = 0; ks < KSTEPS; ++ks) {
            v2f a = *(const v2f*)(arow + c0 + ks * 4 + kk0);  // ds_load_b64, aligned
            if (ALONG_H)
                acc = __builtin_amdgcn_wmma_f32_16x16x4_f32(
                    false, wf[ks], false, a, (short)0, acc, false, false);
            else
                acc = __builtin_amdgcn_wmma_f32_16x16x4_f32(
                    false, a, false, wf[ks], (short)0, acc, false, false);
        }
        if (!ALONG_H) {
            // D: rows = flat rows, cols = w   (stores coalesced over idx)
            float* dst = out + (size_t)blockIdx.x * (16 * 512);
#pragma unroll
            for (int r = 0; r < 8; ++r)
                dst[(size_t)(r + 8 * half) * 512 + c0 + idx] = acc[r];
        } else {
            // D: rows = h outputs, cols = w   (stores coalesced over idx)
            float* dst = out + (size_t)blockIdx.y * IMG + blockIdx.x * 16;
#pragma unroll
            for (int r = 0; r < 8; ++r)
                dst[(size_t)(c0 + r + 8 * half) * 512 + idx] = acc[r];
        }
    }
}

// ---------------------------------------------------------------------------
// Channel pass: radius-20 reflect blur over size-3 axis folds to a 3x3 mix.
// In-place (each thread owns one (n,h,w) column of 3 values).
// ---------------------------------------------------------------------------
__global__ __launch_bounds__(256) void blur_c_inplace(float* data,
                                                      const int* __restrict__ k_size) {
    __shared__ float s_w[KLEN];
    __shared__ float s_M[9];
    __shared__ float s_norm;
    const int tid = threadIdx.x;
    const float sigma = (float)k_size[0];

    if (tid < KLEN) {
        float t = (float)(tid - RADIUS) / sigma;
        s_w[tid] = __expf(-0.5f * t * t);
    }
    __syncthreads();
    if (tid == 0) {
        float s = 0.0f;
        for (int k = 0; k < KLEN; ++k) s += s_w[k];
        s_norm = 1.0f / s;
    }
    __syncthreads();
    if (tid < KLEN) s_w[tid] *= s_norm;
    __syncthreads();
    if (tid < 9) {
        int c = tid / 3, cp = tid % 3;
        float s = 0.0f;
        for (int k = 0; k < KLEN; ++k)
            if (reflect_gen(c + k - RADIUS, 3) == cp) s += s_w[k];
        s_M[tid] = s;
    }
    __syncthreads();

    size_t p = (size_t)blockIdx.x * 256 + tid;   // over 32*512*512 positions
    size_t n = p >> 18, hw = p & (IMG - 1);
    float* base = data + n * NSTR + hw;
    float x0 = base[0], x1 = base[CSTR], x2 = base[2 * CSTR];
    float y0 = s_M[0] * x0 + s_M[1] * x1 + s_M[2] * x2;
    float y1 = s_M[3] * x0 + s_M[4] * x1 + s_M[5] * x2;
    float y2 = s_M[6] * x0 + s_M[7] * x1 + s_M[8] * x2;
    base[0] = y0; base[CSTR] = y1; base[2 * CSTR] = y2;
}

// ---------------------------------------------------------------------------
// Batch pass: radius-20 reflect blur over size-32 axis folds to a 32x32 mix.
// In-place (each thread owns one (c,h,w) column of 32 values).
// ---------------------------------------------------------------------------
__global__ __launch_bounds__(256) void blur_n_inplace(float* data,
                                                      const int* __restrict__ k_size) {
    __shared__ float s_w[KLEN];
    __shared__ float s_M[32 * 32];
    __shared__ float s_norm;
    const int tid = threadIdx.x;
    const float sigma = (float)k_size[0];

    if (tid < KLEN) {
        float t = (float)(tid - RADIUS) / sigma;
        s_w[tid] = __expf(-0.5f * t * t);
    }
    __syncthreads();
    if (tid == 0) {
        float s = 0.0f;
        for (int k = 0; k < KLEN; ++k) s += s_w[k];
        s_norm = 1.0f / s;
    }
    __syncthreads();
    if (tid < KLEN) s_w[tid] *= s_norm;
    __syncthreads();
    for (int e = tid; e < 1024; e += 256) {
        int i = e >> 5, ip = e & 31;
        float s = 0.0f;
        for (int k = 0; k < KLEN; ++k)
            if (reflect_gen(i + k - RADIUS, 32) == ip) s += s_w[k];
        s_M[e] = s;
    }
    __syncthreads();

    size_t p = (size_t)blockIdx.x * 256 + tid;   // over 3*512*512 positions
    float acc[32];
#pragma unroll
    for (int i = 0; i < 32; ++i) acc[i] = 0.0f;
    for (int ip = 0; ip < 32; ++ip) {
        float v = data[(size_t)ip * NSTR + p];
#pragma unroll
        for (int i = 0; i < 32; ++i) acc[i] += s_M[i * 32 + ip] * v;
    }
#pragma unroll
    for (int i = 0; i < 32; ++i) data[(size_t)i * NSTR + p] = acc[i];
}

extern "C" void kernel_launch(void* const* d_in, const int* in_sizes, int n_in,
                              void* d_out, int out_size, void* d_ws, size_t ws_size,
                              hipStream_t stream) {
    const float* x  = (const float*)d_in[0];
    const int*   ks = (const int*)d_in[1];
    float* out = (float*)d_out;
    float* tmp = (float*)d_ws;   // needs 32*3*512*512*4 = 100,663,296 B of scratch

    dim3 blk(256);
    // pass W: x -> tmp      (3072 row-groups of 16 rows x 512)
    blur512_wmma<false><<<dim3(3072), blk, 0, stream>>>(x, tmp, ks);
    // pass H: tmp -> out    (32 w-groups x 96 images)
    blur512_wmma<true><<<dim3(32, 96), blk, 0, stream>>>(tmp, out, ks);
    // pass C (folded 3x3), in place on out
    blur_c_inplace<<<dim3((32 * IMG) / 256), blk, 0, stream>>>(out, ks);
    // pass N (folded 32x32), in place on out
    blur_n_inplace<<<dim3(NSTR / 256), blk, 0, stream>>>(out, ks);
}